// WassersteinLoss_10471130268052
// MI455X (gfx1250) — compile-verified
//
#include <hip/hip_runtime.h>
#include <hip/hip_bf16.h>

// ---------------------------------------------------------------------------
// Debiased Sinkhorn divergence (geomloss 'sinkhorn', p=1, blur=0.01, debias)
// N = M = 6144, d = 3, 32 averaged iterations + 1 extrapolation.
//
// MI455X strategy: compute-bound recomputation of cost tiles (zero HBM
// traffic for the 3x151MB cost matrices). V_WMMA_F32_16X16X4_F32 computes
// dot' = -2 p.q + |q|^2 directly (B columns pre-scaled, A K=4 slot = 1.0),
// so d^2 = dot' + |p|^2 is one add. Streaming base-2 logsumexp, chunked
// 4 tiles at a time: 1 v_sqrt + 1.5 v_exp per element, raw HW trans ops.
// ---------------------------------------------------------------------------

typedef __attribute__((ext_vector_type(2))) float v2f;
typedef __attribute__((ext_vector_type(8))) float v8f;

#define NPTS       6144
#define EPS        0.01f
#define INV_EPS    100.0f
#define LOG2E      1.4426950408889634f
#define LN2        0.6931471805599453f
#define NEG_LOG_N  (-8.723231275f)           /* -ln(6144) */
#define NEG_SCALE  (-144.26950408889634f)    /* -INV_EPS * LOG2E */
#define N_ITERS    32

// workspace layout (floats): f,g,px,py, ft,gt,pxt,pyt  (each NPTS)
#define OF_F   0
#define OF_G   (1*NPTS)
#define OF_PX  (2*NPTS)
#define OF_PY  (3*NPTS)
#define OF_PXT (6*NPTS)
#define OF_FT  (4*NPTS)
#define OF_GT  (5*NPTS)
#define OF_PYT (7*NPTS)

__device__ __forceinline__ float hw_exp2(float v) { return __builtin_amdgcn_exp2f(v); }
__device__ __forceinline__ float hw_log2(float v) { return __builtin_amdgcn_logf(v); }
__device__ __forceinline__ float hw_sqrt(float v) { return __builtin_amdgcn_sqrtf(v); }

// --------------------------------------------------------------------------
__global__ void sink_init_kernel(float* ws) {
    int i = blockIdx.x * blockDim.x + threadIdx.x;
    if (i < 4 * NPTS) ws[i] = 0.0f;
}

// --------------------------------------------------------------------------
// One softmin sweep for all 4 problems.
//   out[i] = -eps * LSE_j( h[j] - ||P_i - Q_j|| / eps )
// grid = (48, 4): blockIdx.y selects (P,Q,pot,out); 8 waves/block, each wave
// owns a 16-row strip, loops over 96 chunks of 4 column tiles.
__global__ __launch_bounds__(256) void sink_sweep_kernel(const float* __restrict__ x,
                                                         const float* __restrict__ y,
                                                         float* __restrict__ ws) {
    // LDS: float4 qb[6144] = (-2q0,-2q1,-2q2,|q|^2), then float hh[6144]
    extern __shared__ char smem[];
    float4* qb = (float4*)smem;
    float*  hh = (float*)(smem + NPTS * sizeof(float4));

    const int sweep = blockIdx.y;    // 0:f<-g  1:g<-f  2:px  3:py
    const float* P   = (sweep == 0 || sweep == 2) ? x : y;
    const float* Q   = (sweep == 0 || sweep == 3) ? y : x;
    const float* pot;
    float*       out;
    if      (sweep == 0) { pot = ws + OF_G;  out = ws + OF_FT;  }
    else if (sweep == 1) { pot = ws + OF_F;  out = ws + OF_GT;  }
    else if (sweep == 2) { pot = ws + OF_PX; out = ws + OF_PXT; }
    else                 { pot = ws + OF_PY; out = ws + OF_PYT; }

    // ---- stage Q columns (pre-scaled for the WMMA) + potentials ----------
    for (int j = threadIdx.x; j < NPTS; j += 256) {
        float q0 = Q[3 * j + 0];
        float q1 = Q[3 * j + 1];
        float q2 = Q[3 * j + 2];
        float qn = fmaf(q0, q0, fmaf(q1, q1, q2 * q2));
        qb[j] = make_float4(-2.0f * q0, -2.0f * q1, -2.0f * q2, qn);
        hh[j] = (NEG_LOG_N + pot[j] * INV_EPS) * LOG2E;   // base-2 domain
    }
    __syncthreads();

    const int lane = threadIdx.x & 31;
    const int wave = threadIdx.x >> 5;                 // 0..7
    const int strip = blockIdx.x * 8 + wave;           // 0..383
    const int rowbase = strip * 16;
    const bool hi = (lane >= 16);

    // ---- loop-invariant A operand: rows (p0,p1,p2,1), K padded to 4 ------
    // A layout: lane L holds row m = L%16; VGPR0/1 = K{0,1} (L<16) or K{2,3}
    const int rA = rowbase + (lane & 15);
    v2f a;
    a.x = hi ? P[3 * rA + 2] : P[3 * rA + 0];
    a.y = hi ? 1.0f          : P[3 * rA + 1];

    // |P_row|^2 for the 8 rows this lane accumulates (D layout: VGPR v ->
    // row v (lanes 0-15) or v+8 (lanes 16-31), col = lane%16).
    float p2r[8];
#pragma unroll
    for (int v = 0; v < 8; ++v) {
        int r = rowbase + v + (hi ? 8 : 0);
        float c0 = P[3 * r + 0], c1 = P[3 * r + 1], c2 = P[3 * r + 2];
        p2r[v] = fmaf(c0, c0, fmaf(c1, c1, c2 * c2));
    }

    // B-operand LDS base: lanes<16 read K{0,1} (bytes 0..7 of the column
    // row), lanes>=16 read K{2,3} (bytes 8..15) -> select by addressing.
    const float* qbf = (const float*)qb;
    const int col0 = lane & 15;                        // column within tile
    const float2* bptr = (const float2*)&qbf[4 * col0 + (hi ? 2 : 0)];
    const float*  hptr = &hh[col0];

    // streaming LSE state (base-2): running max + rescaled sum, per row
    float mm[8], ss[8];
#pragma unroll
    for (int v = 0; v < 8; ++v) { mm[v] = -3.0e38f; ss[v] = 0.0f; }

    const v8f czero = {};

    // ---- main loop: chunks of 4 column tiles (64 columns) ----------------
    for (int j0 = 0; j0 < NPTS; j0 += 64) {
        float tt[4][8];   // base-2 logits, 4 tiles x 8 rows per lane
#pragma unroll
        for (int k = 0; k < 4; ++k) {
            const int jt = j0 + 16 * k;
            float2 b2 = bptr[2 * jt];                  // ds_load_b64
            float  hq = hptr[jt];                      // ds_load_b32
            v2f b; b.x = b2.x; b.y = b2.y;

            // 16x16 tile of dot' = -2 p.q + |q|^2, one instruction
            v8f dot = __builtin_amdgcn_wmma_f32_16x16x4_f32(
                false, a, false, b, (short)0, czero, false, false);

#pragma unroll
            for (int v = 0; v < 8; ++v) {
                float d2 = dot[v] + p2r[v];            // = |p-q|^2
                float c  = hw_sqrt(fmaxf(d2, 1e-12f)); // ||P_i - Q_j||
                tt[k][v] = fmaf(c, NEG_SCALE, hq);     // (h - C/eps)*log2e
            }
        }
        // chunk-local max, then one merge into the running (m, s)
#pragma unroll
        for (int v = 0; v < 8; ++v) {
            float cm = fmaxf(fmaxf(tt[0][v], tt[1][v]), fmaxf(tt[2][v], tt[3][v]));
            float sc = hw_exp2(tt[0][v] - cm) + hw_exp2(tt[1][v] - cm)
                     + hw_exp2(tt[2][v] - cm) + hw_exp2(tt[3][v] - cm);
            float mo = mm[v];
            float mn = fmaxf(mo, cm);
            ss[v] = fmaf(ss[v], hw_exp2(mo - mn), sc * hw_exp2(cm - mn));
            mm[v] = mn;
        }
    }

    // ---- combine (m,s) across the 16 lanes sharing each row set ----------
#pragma unroll
    for (int d = 1; d <= 8; d <<= 1) {
#pragma unroll
        for (int v = 0; v < 8; ++v) {
            float mo = __shfl_xor(mm[v], d, 32);
            float so = __shfl_xor(ss[v], d, 32);
            float mn = fmaxf(mm[v], mo);
            ss[v] = fmaf(ss[v], hw_exp2(mm[v] - mn), so * hw_exp2(mo - mn));
            mm[v] = mn;
        }
    }

    if ((lane & 15) == 0) {
        int rowoff = rowbase + (hi ? 8 : 0);
#pragma unroll
        for (int v = 0; v < 8; ++v) {
            // -eps * ln(sum exp) = -eps * ln2 * (m2 + log2(s))
            out[rowoff + v] = -EPS * LN2 * (mm[v] + hw_log2(ss[v]));
        }
    }
}

// --------------------------------------------------------------------------
__global__ void sink_update_kernel(float* ws) {
    int i = blockIdx.x * blockDim.x + threadIdx.x;
    if (i < NPTS) {
#pragma unroll
        for (int a = 0; a < 4; ++a) {
            float cur = ws[a * NPTS + i];
            float tgt = ws[(4 + a) * NPTS + i];
            ws[a * NPTS + i] = 0.5f * (cur + tgt);
        }
    }
}

// --------------------------------------------------------------------------
__global__ void sink_reduce_kernel(const float* __restrict__ ws,
                                   float* __restrict__ out) {
    __shared__ float red[8];
    float acc = 0.0f;
    for (int i = threadIdx.x; i < NPTS; i += 256) {
        acc += (ws[OF_FT + i] - ws[OF_PXT + i]) + (ws[OF_GT + i] - ws[OF_PYT + i]);
    }
#pragma unroll
    for (int d = 16; d >= 1; d >>= 1) acc += __shfl_xor(acc, d, 32);
    if ((threadIdx.x & 31) == 0) red[threadIdx.x >> 5] = acc;
    __syncthreads();
    if (threadIdx.x == 0) {
        float t = 0.0f;
#pragma unroll
        for (int w = 0; w < 8; ++w) t += red[w];
        out[0] = t * (1.0f / (float)NPTS);   // a_i = b_j = 1/N
    }
}

// --------------------------------------------------------------------------
extern "C" void kernel_launch(void* const* d_in, const int* in_sizes, int n_in,
                              void* d_out, int out_size, void* d_ws, size_t ws_size,
                              hipStream_t stream) {
    (void)in_sizes; (void)n_in; (void)out_size; (void)ws_size;
    const float* x = (const float*)d_in[0];   // [6144,3]
    const float* y = (const float*)d_in[1];   // [6144,3]
    float* ws  = (float*)d_ws;                // 8*6144 floats
    float* out = (float*)d_out;               // scalar

    const dim3 sweepGrid(48, 4);
    const size_t ldsBytes = NPTS * (sizeof(float4) + sizeof(float));  // 120KB

    sink_init_kernel<<<(4 * NPTS + 255) / 256, 256, 0, stream>>>(ws);

    for (int it = 0; it < N_ITERS; ++it) {
        sink_sweep_kernel<<<sweepGrid, 256, ldsBytes, stream>>>(x, y, ws);
        sink_update_kernel<<<(NPTS + 255) / 256, 256, 0, stream>>>(ws);
    }
    // final extrapolation (no averaging): targets hold f_fin,g_fin,px_fin,py_fin
    sink_sweep_kernel<<<sweepGrid, 256, ldsBytes, stream>>>(x, y, ws);

    sink_reduce_kernel<<<1, 256, 0, stream>>>(ws, out);
}